// ProjectionDepthSimilarityLoss_56057913147750
// MI455X (gfx1250) — compile-verified
//
#include <hip/hip_runtime.h>

#define IMH 480
#define IMW 640
#define NPIX (IMH * IMW)

typedef __attribute__((ext_vector_type(2))) float v2f;
typedef __attribute__((ext_vector_type(8))) float v8f;

// ---------------------------------------------------------------------------
// Wave32 reduction via V_WMMA_F32_16X16X4_F32:
//   A(16x4): lane's partial in one slot, 0 elsewhere; B(4x16) = ones
//   => D[m][n] = rowsum(A[m][:]) ; lane L holds 8 distinct rows in its v8f D;
//   lanes L and L^16 together cover all 16 rows exactly once.
// EXEC must be all 1s at the call site (guaranteed by callers).
// ---------------------------------------------------------------------------
__device__ __forceinline__ float wave_reduce_wmma(float p) {
  v2f a; a[0] = p;    a[1] = 0.0f;
  v2f b; b[0] = 1.0f; b[1] = 1.0f;
  v8f c = {};
  v8f d = __builtin_amdgcn_wmma_f32_16x16x4_f32(
      /*neg_a=*/false, a, /*neg_b=*/false, b,
      /*c_mod=*/(short)0, c, /*reuse_a=*/false, /*reuse_b=*/false);
  float t = ((d[0] + d[1]) + (d[2] + d[3])) + ((d[4] + d[5]) + (d[6] + d[7]));
  t += __shfl_xor(t, 16, 32);   // pair the two half-wave row groups
  return t;                      // full wave sum in every lane
}

// ---------------------------------------------------------------------------
// Per-batch transform composition:
//   uvd = depth * (ray @ G) + c
//   G0 = R0 @ R1^T @ K^T ; c0 = (t1 - t0 @ (R0@R1^T)) @ K^T
//   dir1 uses Q = (R0@R1^T)^T.
// Output layout per batch: G0[9], c0[3], G1[9], c1[3]  (24 floats)
// ---------------------------------------------------------------------------
__global__ void precompute_mats(const float* __restrict__ R0g, const float* __restrict__ t0g,
                                const float* __restrict__ R1g, const float* __restrict__ t1g,
                                const float* __restrict__ Kg, float* __restrict__ mats, int B) {
  int b = threadIdx.x;
  if (b >= B) return;
  float r0[9], r1[9], k[9], t0[3], t1[3], P[9];
  #pragma unroll
  for (int i = 0; i < 9; ++i) { r0[i] = R0g[b * 9 + i]; r1[i] = R1g[b * 9 + i]; k[i] = Kg[i]; }
  #pragma unroll
  for (int i = 0; i < 3; ++i) { t0[i] = t0g[b * 3 + i]; t1[i] = t1g[b * 3 + i]; }
  #pragma unroll
  for (int i = 0; i < 3; ++i)
    #pragma unroll
    for (int j = 0; j < 3; ++j)
      P[i * 3 + j] = r0[i * 3 + 0] * r1[j * 3 + 0] + r0[i * 3 + 1] * r1[j * 3 + 1] +
                     r0[i * 3 + 2] * r1[j * 3 + 2];
  float* o = mats + b * 24;
  // G0 = P @ K^T
  #pragma unroll
  for (int i = 0; i < 3; ++i)
    #pragma unroll
    for (int j = 0; j < 3; ++j)
      o[i * 3 + j] = P[i * 3 + 0] * k[j * 3 + 0] + P[i * 3 + 1] * k[j * 3 + 1] +
                     P[i * 3 + 2] * k[j * 3 + 2];
  // c0 = (t1 - t0 @ P) @ K^T
  float w0[3];
  #pragma unroll
  for (int kk = 0; kk < 3; ++kk)
    w0[kk] = t1[kk] - (t0[0] * P[0 * 3 + kk] + t0[1] * P[1 * 3 + kk] + t0[2] * P[2 * 3 + kk]);
  #pragma unroll
  for (int j = 0; j < 3; ++j)
    o[9 + j] = w0[0] * k[j * 3 + 0] + w0[1] * k[j * 3 + 1] + w0[2] * k[j * 3 + 2];
  // G1 = P^T @ K^T
  #pragma unroll
  for (int i = 0; i < 3; ++i)
    #pragma unroll
    for (int j = 0; j < 3; ++j)
      o[12 + i * 3 + j] = P[0 * 3 + i] * k[j * 3 + 0] + P[1 * 3 + i] * k[j * 3 + 1] +
                          P[2 * 3 + i] * k[j * 3 + 2];
  // c1 = (t0 - t1 @ P^T) @ K^T ; (t1 @ P^T)_k = sum_i t1_i * P[k][i]
  float w1[3];
  #pragma unroll
  for (int kk = 0; kk < 3; ++kk)
    w1[kk] = t0[kk] - (t1[0] * P[kk * 3 + 0] + t1[1] * P[kk * 3 + 1] + t1[2] * P[kk * 3 + 2]);
  #pragma unroll
  for (int j = 0; j < 3; ++j)
    o[21 + j] = w1[0] * k[j * 3 + 0] + w1[1] * k[j * 3 + 1] + w1[2] * k[j * 3 + 2];
}

// ---------------------------------------------------------------------------
// Project + bilinear border sample + |d - s| for one pixel/direction.
// ---------------------------------------------------------------------------
__device__ __forceinline__ float sample_diff(float dep, float rx, float ry, float rz,
                                             const float* __restrict__ g,
                                             const float* __restrict__ c,
                                             const float* __restrict__ img) {
  float rm0 = rx * g[0] + ry * g[3] + rz * g[6];
  float rm1 = rx * g[1] + ry * g[4] + rz * g[7];
  float rm2 = rx * g[2] + ry * g[5] + rz * g[8];
  float u0 = fmaf(dep, rm0, c[0]);
  float u1 = fmaf(dep, rm1, c[1]);
  float u2 = fmaf(dep, rm2, c[2]);
  float denom = fmaxf(u2, 0.0f) + 1e-12f;
  float inv = 1.0f / denom;
  float gx = 2.0f * (u0 * inv * (1.0f / (IMW - 1)) - 0.5f);
  float gy = 2.0f * (u1 * inv * (1.0f / (IMH - 1)) - 0.5f);
  float x = fminf(fmaxf(((gx + 1.0f) * (float)IMW - 1.0f) * 0.5f, 0.0f), (float)(IMW - 1));
  float y = fminf(fmaxf(((gy + 1.0f) * (float)IMH - 1.0f) * 0.5f, 0.0f), (float)(IMH - 1));
  float x0f = floorf(x), y0f = floorf(y);
  float wx = x - x0f, wy = y - y0f;
  int x0 = (int)x0f, y0 = (int)y0f;
  int x1 = min(x0 + 1, IMW - 1), y1 = min(y0 + 1, IMH - 1);
  float v00 = img[y0 * IMW + x0];
  float v01 = img[y0 * IMW + x1];
  float v10 = img[y1 * IMW + x0];
  float v11 = img[y1 * IMW + x1];
  float top = fmaf(wx, v01 - v00, v00);
  float bot = fmaf(wx, v11 - v10, v10);
  float s = fmaf(wy, bot - top, top);
  return fabsf(u2 - s);
}

// ---------------------------------------------------------------------------
// Main kernel: grid = (300, B), 256 threads, 4 pixels/thread (exact tiling,
// EXEC stays all-ones; 128-bit loads for depth and rays).
// ---------------------------------------------------------------------------
__global__ void __launch_bounds__(256)
proj_loss_main(const float* __restrict__ depth0, const float* __restrict__ depth1,
               const float* __restrict__ ray, const float* __restrict__ mats,
               float* __restrict__ partials) {
  const int b = blockIdx.y;                 // uniform -> scalar loads below
  const float* M = mats + b * 24;
  float g0[9], c0[3], g1[9], c1[3];
  #pragma unroll
  for (int i = 0; i < 9; ++i) { g0[i] = M[i]; g1[i] = M[12 + i]; }
  #pragma unroll
  for (int i = 0; i < 3; ++i) { c0[i] = M[9 + i]; c1[i] = M[21 + i]; }

  const int tid = threadIdx.x;
  const int n0 = blockIdx.x * 1024 + tid * 4;
  const size_t base = (size_t)b * NPIX;
  const float4 dv0 = *(const float4*)(depth0 + base + n0);
  const float4 dv1 = *(const float4*)(depth1 + base + n0);
  const float4* rp = (const float4*)(ray + (size_t)3 * n0);
  float4 rv0 = rp[0], rv1 = rp[1], rv2 = rp[2];
  float rf[12];
  *(float4*)(rf + 0) = rv0; *(float4*)(rf + 4) = rv1; *(float4*)(rf + 8) = rv2;
  const float* d0f = (const float*)&dv0;
  const float* d1f = (const float*)&dv1;
  const float* img1 = depth1 + base;
  const float* img0 = depth0 + base;

  float acc = 0.0f;
  #pragma unroll
  for (int i = 0; i < 4; ++i) {
    float rx = rf[3 * i + 0], ry = rf[3 * i + 1], rz = rf[3 * i + 2];
    acc += sample_diff(d0f[i], rx, ry, rz, g0, c0, img1);   // frame0 -> frame1
    acc += sample_diff(d1f[i], rx, ry, rz, g1, c1, img0);   // frame1 -> frame0
  }

  float wsum = wave_reduce_wmma(acc);       // EXEC all-ones here
  __shared__ float lds[8];
  const int lane = tid & 31, wid = tid >> 5;
  if (lane == 0) lds[wid] = wsum;
  __syncthreads();
  if (tid == 0) {
    float s = 0.0f;
    #pragma unroll
    for (int w = 0; w < 8; ++w) s += lds[w];
    partials[blockIdx.y * gridDim.x + blockIdx.x] = s;
  }
}

// ---------------------------------------------------------------------------
// Deterministic final reduction: single block, fixed accumulation order.
// ---------------------------------------------------------------------------
__global__ void __launch_bounds__(256)
final_reduce(const float* __restrict__ partials, int n, float* __restrict__ out, float scale) {
  float acc = 0.0f;
  for (int i = threadIdx.x; i < n; i += 256) acc += partials[i];
  // loop reconverges: EXEC all-ones for the WMMA reduce
  float wsum = wave_reduce_wmma(acc);
  __shared__ float lds[8];
  const int lane = threadIdx.x & 31, wid = threadIdx.x >> 5;
  if (lane == 0) lds[wid] = wsum;
  __syncthreads();
  if (threadIdx.x == 0) {
    float s = 0.0f;
    #pragma unroll
    for (int w = 0; w < 8; ++w) s += lds[w];
    out[0] = s * scale;
  }
}

extern "C" void kernel_launch(void* const* d_in, const int* in_sizes, int n_in,
                              void* d_out, int out_size, void* d_ws, size_t ws_size,
                              hipStream_t stream) {
  const float* depth0 = (const float*)d_in[0];
  const float* depth1 = (const float*)d_in[1];
  const float* R0 = (const float*)d_in[2];
  const float* t0 = (const float*)d_in[3];
  const float* R1 = (const float*)d_in[4];
  const float* t1 = (const float*)d_in[5];
  const float* Km = (const float*)d_in[6];
  const float* ray = (const float*)d_in[7];
  const int B = in_sizes[0] / NPIX;                // 16

  float* mats = (float*)d_ws;                      // 24*B floats
  float* partials = (float*)((char*)d_ws + 4096);  // 300*B floats

  precompute_mats<<<1, 32, 0, stream>>>(R0, t0, R1, t1, Km, mats, B);

  dim3 grid(NPIX / 1024, B);                       // (300, 16), exact tiling
  proj_loss_main<<<grid, 256, 0, stream>>>(depth0, depth1, ray, mats, partials);

  const int nblocks = (NPIX / 1024) * B;
  final_reduce<<<1, 256, 0, stream>>>(partials, nblocks, (float*)d_out,
                                      1.0f / ((float)B * (float)NPIX));
}